// DirGraphSAGE_Rossi_83408264888607
// MI455X (gfx1250) — compile-verified
//
#include <hip/hip_runtime.h>

#define NN 50000
#define EE 800000
#define DD 128
#define K3 384

typedef __attribute__((ext_vector_type(2))) float v2f;
typedef __attribute__((ext_vector_type(8))) float v8f;

// ---------------------------------------------------------------------------
// Degree accumulation: one thread per edge, float atomics into L2.
// ---------------------------------------------------------------------------
__global__ __launch_bounds__(256) void deg_kernel(
    const int* __restrict__ src, const int* __restrict__ dst,
    float* __restrict__ deg_out, float* __restrict__ deg_in, int E) {
  int e = blockIdx.x * blockDim.x + threadIdx.x;
  if (e >= E) return;
  atomicAdd(&deg_out[src[e]], 1.0f);
  atomicAdd(&deg_in[dst[e]], 1.0f);
}

// deg -> 1 / max(deg, 1)   (covers both deg arrays in one contiguous pass)
__global__ __launch_bounds__(256) void inv_kernel(float* __restrict__ deg, int n) {
  int i = blockIdx.x * blockDim.x + threadIdx.x;
  if (i >= n) return;
  deg[i] = 1.0f / fmaxf(deg[i], 1.0f);
}

// ---------------------------------------------------------------------------
// Edge aggregation, both directions in one pass.
// 32 threads per edge; each thread owns a float4 feature slice.
//   fwd[dst] += x[src] * inv_out[src]
//   bwd[src] += x[dst] * inv_in[dst]
// ---------------------------------------------------------------------------
__global__ __launch_bounds__(256) void agg_kernel(
    const float* __restrict__ x,
    const int* __restrict__ src, const int* __restrict__ dst,
    const float* __restrict__ inv_out, const float* __restrict__ inv_in,
    float* __restrict__ fwd, float* __restrict__ bwd, int E) {
  long long tid = (long long)blockIdx.x * blockDim.x + threadIdx.x;
  int e = (int)(tid >> 5);
  if (e >= E) return;
  int f4 = ((int)tid & 31) * 4;

  int s = src[e];
  int d = dst[e];
  float io = inv_out[s];
  float ii = inv_in[d];

  float4 xs = *(const float4*)(x + (size_t)s * DD + f4);
  float4 xd = *(const float4*)(x + (size_t)d * DD + f4);

  float* fp = fwd + (size_t)d * DD + f4;
  float* bp = bwd + (size_t)s * DD + f4;
  atomicAdd(fp + 0, xs.x * io);
  atomicAdd(fp + 1, xs.y * io);
  atomicAdd(fp + 2, xs.z * io);
  atomicAdd(fp + 3, xs.w * io);
  atomicAdd(bp + 0, xd.x * ii);
  atomicAdd(bp + 1, xd.y * ii);
  atomicAdd(bp + 2, xd.z * ii);
  atomicAdd(bp + 3, xd.w * ii);
}

// ---------------------------------------------------------------------------
// Fused concat + GEMM + bias (+ReLU) using V_WMMA_F32_16X16X4_F32.
//   out[16 rows x 128 cols] = [x | fwd | bwd](16 x 384) @ W(384 x 128) + b
// One block = 16 output rows; 8 waves, one 16x16 output tile per wave.
// A tile staged in LDS (reads the three source buffers directly, no
// materialized concat). B fed from global (W is hot in L1/L2).
// ---------------------------------------------------------------------------
__global__ __launch_bounds__(256) void gemm_wmma_kernel(
    const float* __restrict__ x, const float* __restrict__ fwd,
    const float* __restrict__ bwd, const float* __restrict__ W,
    const float* __restrict__ bias, float* __restrict__ out, int do_relu) {
  __shared__ float As[16][K3 + 4];  // +4 pad to stagger LDS banks

  const int rowBase = blockIdx.x * 16;
  const int tid = threadIdx.x;
  const int wave = tid >> 5;     // 0..7 -> 16-col tile
  const int lane = tid & 31;
  const int hl = lane >> 4;      // lane half: selects K {0,1} vs {2,3}
  const int l16 = lane & 15;

  // Stage A = [x | fwd | bwd] rows rowBase..rowBase+15 into LDS.
  for (int i = tid; i < 16 * K3; i += 256) {
    int r = i / K3;
    int k = i - r * K3;
    size_t node = (size_t)(rowBase + r);
    float v;
    if (k < DD)            v = x[node * DD + k];
    else if (k < 2 * DD)   v = fwd[node * DD + (k - DD)];
    else                   v = bwd[node * DD + (k - 2 * DD)];
    As[r][k] = v;
  }
  __syncthreads();

  const int colBase = wave * 16;
  const int col = colBase + l16;

  v8f acc = {};
  for (int k0 = 0; k0 < K3; k0 += 4) {
    // A 16x4 fp32: lanes 0-15 / 16-31 hold K={0,1} / K={2,3} for row M=l16
    v2f a;
    a.x = As[l16][k0 + 2 * hl + 0];
    a.y = As[l16][k0 + 2 * hl + 1];
    // B 4x16 fp32: row K striped across the 2 VGPRs, cols across lanes
    v2f b;
    b.x = W[(size_t)(k0 + 2 * hl + 0) * DD + col];
    b.y = W[(size_t)(k0 + 2 * hl + 1) * DD + col];
    acc = __builtin_amdgcn_wmma_f32_16x16x4_f32(
        /*neg_a=*/false, a, /*neg_b=*/false, b,
        /*c_mod=*/(short)0, acc, /*reuse_a=*/false, /*reuse_b=*/false);
  }

  // Epilogue: C/D layout — VGPR r: lanes 0-15 -> M=r, lanes 16-31 -> M=r+8.
  const float bb = bias[col];
#pragma unroll
  for (int r = 0; r < 8; ++r) {
    float v = acc[r] + bb;
    if (do_relu) v = fmaxf(v, 0.0f);
    out[(size_t)(rowBase + r + 8 * hl) * DD + col] = v;
  }
}

// ---------------------------------------------------------------------------
// Orchestration.  ws layout (floats):
//   [0, N)           deg_out -> inv_out
//   [N, 2N)          deg_in  -> inv_in
//   [2N, 2N+ND)      xA   (layer0 output)
//   ...              xB   (layer1 output)
//   ...              fwd accumulator
//   ...              bwd accumulator   (fwd/bwd contiguous -> single memset)
// ---------------------------------------------------------------------------
extern "C" void kernel_launch(void* const* d_in, const int* in_sizes, int n_in,
                              void* d_out, int out_size, void* d_ws, size_t ws_size,
                              hipStream_t stream) {
  const float* x  = (const float*)d_in[0];
  const float* W0 = (const float*)d_in[1];
  const float* b0 = (const float*)d_in[2];
  const float* W1 = (const float*)d_in[3];
  const float* b1 = (const float*)d_in[4];
  const float* W2 = (const float*)d_in[5];
  const float* b2 = (const float*)d_in[6];
  const int* src  = (const int*)d_in[7];
  const int* dst  = (const int*)d_in[8];
  float* out = (float*)d_out;

  float* ws      = (float*)d_ws;
  float* deg_out = ws;                       // N
  float* deg_in  = ws + NN;                  // N
  float* xA      = ws + 2 * (size_t)NN;      // N*D
  float* xB      = xA + (size_t)NN * DD;     // N*D
  float* fwd     = xB + (size_t)NN * DD;     // N*D
  float* bwd     = fwd + (size_t)NN * DD;    // N*D

  // Degrees (once).
  hipMemsetAsync(deg_out, 0, 2 * (size_t)NN * sizeof(float), stream);
  deg_kernel<<<(EE + 255) / 256, 256, 0, stream>>>(src, dst, deg_out, deg_in, EE);
  inv_kernel<<<(2 * NN + 255) / 256, 256, 0, stream>>>(deg_out, 2 * NN);

  const float* Ws[3]  = {W0, W1, W2};
  const float* bs[3]  = {b0, b1, b2};
  float*       dsts[3] = {xA, xB, out};

  const float* cur = x;
  const long long agg_threads = (long long)EE * 32;
  const int agg_blocks = (int)((agg_threads + 255) / 256);

  for (int l = 0; l < 3; ++l) {
    hipMemsetAsync(fwd, 0, 2 * (size_t)NN * DD * sizeof(float), stream);
    agg_kernel<<<agg_blocks, 256, 0, stream>>>(cur, src, dst, deg_out, deg_in,
                                               fwd, bwd, EE);
    gemm_wmma_kernel<<<NN / 16, 256, 0, stream>>>(cur, fwd, bwd, Ws[l], bs[l],
                                                  dsts[l], (l < 2) ? 1 : 0);
    cur = dsts[l];
  }
}